// GCN_25194278158726
// MI455X (gfx1250) — compile-verified
//
#include <hip/hip_runtime.h>
#include <hip/hip_bf16.h>
#include <math.h>

typedef float v2f __attribute__((ext_vector_type(2)));
typedef float v8f __attribute__((ext_vector_type(8)));

// ---------------------------------------------------------------------------
// fp32 WMMA GEMM + bias: Y[nrows, ncols] = X[nrows, K] @ W[K, ncols] + bias
// One wave32 computes one 16x16 output tile via V_WMMA_F32_16X16X4_F32.
// A tile (16x4 f32): lanes 0-15 hold M=lane, v0=K+0, v1=K+1;
//                    lanes 16-31 hold M=lane-16, v0=K+2, v1=K+3.
// B tile (4x16 f32): mirrored (lanes 0-15: rows K+0/K+1 at N=lane;
//                    lanes 16-31: rows K+2/K+3 at N=lane-16).
// C/D (16x16 f32):   VGPR r = element (M = r + 8*(lane>=16), N = lane&15).
// nrows, K, ncols must all be multiples of 16/4 (true here: 50000,128,128/64).
// ---------------------------------------------------------------------------
__global__ void gemm_bias_wmma_f32(const float* __restrict__ X,
                                   const float* __restrict__ W,
                                   const float* __restrict__ bias,
                                   float* __restrict__ Y,
                                   int nrows, int K, int ncols) {
  const int wavesPerBlock = blockDim.x >> 5;
  const int wave = blockIdx.x * wavesPerBlock + ((int)threadIdx.x >> 5);
  const int lane = threadIdx.x & 31;
  const int tilesN = ncols >> 4;
  const int tilesM = nrows >> 4;
  if (wave >= tilesM * tilesN) return;          // whole-wave uniform: EXEC stays all-1s
  const int row0 = (wave / tilesN) << 4;
  const int col0 = (wave % tilesN) << 4;
  const int half = lane >> 4;                   // 0: lanes 0-15, 1: lanes 16-31
  const int l16  = lane & 15;

  const float* xrow = X + (size_t)(row0 + l16) * K;
  v8f acc = {};
  for (int k = 0; k < K; k += 4) {
    v2f a, b;
    a.x = xrow[k + 2 * half + 0];
    a.y = xrow[k + 2 * half + 1];
    b.x = W[(size_t)(k + 2 * half + 0) * ncols + col0 + l16];
    b.y = W[(size_t)(k + 2 * half + 1) * ncols + col0 + l16];
    acc = __builtin_amdgcn_wmma_f32_16x16x4_f32(false, a, false, b,
                                                (short)0, acc, false, false);
  }
  const float bv = bias[col0 + l16];
#pragma unroll
  for (int r = 0; r < 8; ++r) {
    const int m = row0 + r + 8 * half;
    Y[(size_t)m * ncols + col0 + l16] = acc[r] + bv;
  }
}

// ---------------------------------------------------------------------------
// Per-call re-init of edge-phase state (ws is poisoned by the harness).
// aggElems >= n always holds here.
// ---------------------------------------------------------------------------
__global__ void init_edge_state(float* __restrict__ smax, float* __restrict__ denom,
                                float* __restrict__ agg, int n, int aggElems) {
  const int i = blockIdx.x * blockDim.x + threadIdx.x;
  if (i < n) { smax[i] = -__builtin_inff(); denom[i] = 0.0f; }
  if (i < aggElems) agg[i] = 0.0f;
}

// Float atomic max via ordered-int trick (native global_atomic_max_i32/min_u32).
__device__ __forceinline__ void atomicMaxF(float* addr, float val) {
  if (val >= 0.0f) atomicMax((int*)addr, __float_as_int(val));
  else             atomicMin((unsigned int*)addr, __float_as_uint(val));
}

// ---------------------------------------------------------------------------
// Pass 1: score[e] = scale * dot(q[dst[e]], k[src[e]]); smax[dst] = max(...)
// One wave per edge; float4 gathers (L2-resident), wave32 xor-shuffle reduce.
// ---------------------------------------------------------------------------
__global__ void edge_score_kernel(const float* __restrict__ q, const float* __restrict__ kk,
                                  const int* __restrict__ src, const int* __restrict__ dst,
                                  float* __restrict__ score, float* __restrict__ smax,
                                  int nedges, int d, float scale) {
  const int wavesPerBlock = blockDim.x >> 5;
  const int e = blockIdx.x * wavesPerBlock + ((int)threadIdx.x >> 5);
  if (e >= nedges) return;
  const int lane = threadIdx.x & 31;
  const int s = src[e];
  const int t = dst[e];
  const float4* q4 = (const float4*)(q + (size_t)t * d);
  const float4* k4 = (const float4*)(kk + (size_t)s * d);
  const int n4 = d >> 2;
  float acc = 0.0f;
  for (int c = lane; c < n4; c += 32) {
    const float4 a = q4[c];
    const float4 b = k4[c];
    acc += a.x * b.x + a.y * b.y + a.z * b.z + a.w * b.w;
  }
#pragma unroll
  for (int off = 16; off > 0; off >>= 1) acc += __shfl_xor(acc, off, 32);
  acc *= scale;
  if (lane == 0) {
    score[e] = acc;
    atomicMaxF(&smax[t], acc);
  }
}

// ---------------------------------------------------------------------------
// Pass 2: ex = exp(score - smax[dst]); denom[dst] += ex; agg[dst,:] += ex*v[src,:]
// (normalization by denom is deferred to finalize — identical to alpha form)
// ---------------------------------------------------------------------------
__global__ void edge_agg_kernel(const float* __restrict__ v, const float* __restrict__ score,
                                const float* __restrict__ smax, const int* __restrict__ src,
                                const int* __restrict__ dst, float* __restrict__ denom,
                                float* __restrict__ agg, int nedges, int d) {
  const int wavesPerBlock = blockDim.x >> 5;
  const int e = blockIdx.x * wavesPerBlock + ((int)threadIdx.x >> 5);
  if (e >= nedges) return;
  const int lane = threadIdx.x & 31;
  const int s = src[e];
  const int t = dst[e];
  const float ex = expf(score[e] - smax[t]);
  if (lane == 0) atomicAdd(&denom[t], ex);
  const float* vp = v + (size_t)s * d;
  float* ap = agg + (size_t)t * d;
  for (int c = lane; c < d; c += 32) atomicAdd(&ap[c], ex * vp[c]);
}

// ---------------------------------------------------------------------------
// out[i] = agg[i] / (denom>0 ? denom : 1) + skip[i]; optional ReLU
// ---------------------------------------------------------------------------
__global__ void finalize_kernel(const float* __restrict__ agg, const float* __restrict__ denom,
                                const float* __restrict__ skip, float* __restrict__ out,
                                int total, int d, int relu) {
  const int i = blockIdx.x * blockDim.x + threadIdx.x;
  if (i >= total) return;
  const int node = i / d;
  float dn = denom[node];
  dn = (dn > 0.0f) ? dn : 1.0f;
  float val = agg[i] / dn + skip[i];
  out[i] = relu ? fmaxf(val, 0.0f) : val;
}

extern "C" void kernel_launch(void* const* d_in, const int* in_sizes, int n_in,
                              void* d_out, int out_size, void* d_ws, size_t ws_size,
                              hipStream_t stream) {
  const int N = 50000, E = 600000, INC = 128, HID = 128, OUTC = 64;

  const float* x  = (const float*)d_in[0];
  const int*   ei = (const int*)d_in[1];
  const int*   src = ei;          // edge_index[0]
  const int*   dst = ei + E;      // edge_index[1]
  // 0:q1 1:k1 2:v1 3:s1 4:q2 5:k2 6:v2 7:s2
  const float* W[8];
  const float* B[8];
  for (int i = 0; i < 8; ++i) {
    W[i] = (const float*)d_in[2 + 2 * i];
    B[i] = (const float*)d_in[3 + 2 * i];
  }

  float* ws    = (float*)d_ws;
  float* q     = ws;
  float* k     = q    + (size_t)N * HID;
  float* v     = k    + (size_t)N * HID;
  float* skip  = v    + (size_t)N * HID;
  float* agg   = skip + (size_t)N * HID;
  float* h     = agg  + (size_t)N * HID;   // layer-1 output
  float* score = h    + (size_t)N * HID;
  float* smax  = score + E;
  float* denom = smax + N;

  const int THREADS = 256;                 // 8 waves per block (wave32)
  const int WPB = THREADS / 32;

  // -------- Layer 1 (d = HID = 128, ReLU) --------
  {
    const int tiles  = (N / 16) * (HID / 16);
    const int blocks = (tiles + WPB - 1) / WPB;
    gemm_bias_wmma_f32<<<blocks, THREADS, 0, stream>>>(x, W[0], B[0], q,    N, INC, HID);
    gemm_bias_wmma_f32<<<blocks, THREADS, 0, stream>>>(x, W[1], B[1], k,    N, INC, HID);
    gemm_bias_wmma_f32<<<blocks, THREADS, 0, stream>>>(x, W[2], B[2], v,    N, INC, HID);
    gemm_bias_wmma_f32<<<blocks, THREADS, 0, stream>>>(x, W[3], B[3], skip, N, INC, HID);

    const int aggE = N * HID;
    init_edge_state<<<(aggE + THREADS - 1) / THREADS, THREADS, 0, stream>>>(smax, denom, agg, N, aggE);

    const int eb = (E + WPB - 1) / WPB;
    edge_score_kernel<<<eb, THREADS, 0, stream>>>(q, k, src, dst, score, smax, E, HID,
                                                  1.0f / sqrtf((float)HID));
    edge_agg_kernel<<<eb, THREADS, 0, stream>>>(v, score, smax, src, dst, denom, agg, E, HID);
    finalize_kernel<<<(aggE + THREADS - 1) / THREADS, THREADS, 0, stream>>>(
        agg, denom, skip, h, aggE, HID, /*relu=*/1);
  }

  // -------- Layer 2 (d = OUTC = 64, no ReLU) --------
  {
    const int tiles  = (N / 16) * (OUTC / 16);
    const int blocks = (tiles + WPB - 1) / WPB;
    gemm_bias_wmma_f32<<<blocks, THREADS, 0, stream>>>(h, W[4], B[4], q,    N, HID, OUTC);
    gemm_bias_wmma_f32<<<blocks, THREADS, 0, stream>>>(h, W[5], B[5], k,    N, HID, OUTC);
    gemm_bias_wmma_f32<<<blocks, THREADS, 0, stream>>>(h, W[6], B[6], v,    N, HID, OUTC);
    gemm_bias_wmma_f32<<<blocks, THREADS, 0, stream>>>(h, W[7], B[7], skip, N, HID, OUTC);

    const int aggE = N * OUTC;
    init_edge_state<<<(aggE + THREADS - 1) / THREADS, THREADS, 0, stream>>>(smax, denom, agg, N, aggE);

    const int eb = (E + WPB - 1) / WPB;
    edge_score_kernel<<<eb, THREADS, 0, stream>>>(q, k, src, dst, score, smax, E, OUTC,
                                                  1.0f / sqrtf((float)OUTC));
    edge_agg_kernel<<<eb, THREADS, 0, stream>>>(v, score, smax, src, dst, denom, agg, E, OUTC);
    finalize_kernel<<<(aggE + THREADS - 1) / THREADS, THREADS, 0, stream>>>(
        agg, denom, skip, (float*)d_out, aggE, OUTC, /*relu=*/0);
  }
}